// ThermalGCN_56762287784217
// MI455X (gfx1250) — compile-verified
//
#include <hip/hip_runtime.h>

// ---------------------------------------------------------------------------
// ThermalGCN on MI455X (gfx1250, wave32).
//
// Pipeline (all on `stream`, serialized by stream order):
//   K1 init:     d_out = 0, deg = 1.0 (self-loop weight)
//   K2 degree:   atomicAdd(deg[col[e]], ew[e])               (per graph)
//   K3 rsqrt:    deg -> dinv = rsqrt(deg) in place
//   K4 scatter:  agg[col] += dinv[row]*ew*dinv[col] * (x[row]@W)  (per graph)
//                Persistent waves, W hoisted to registers, h recomputed per
//                edge (in-dim <= 3). All atomic traffic resident in 192MB L2.
//   K5 finalize: h tile via V_WMMA_F32_16X16X4_F32 (exact fp32), then
//                out = relu(agg + dinv^2 * h + bias). Branch-free full-tile
//                fast path; K-padding done with clamped loads + cndmask
//                (never exec-masked branches).
// ---------------------------------------------------------------------------

typedef __attribute__((ext_vector_type(2))) float v2f;
typedef __attribute__((ext_vector_type(8))) float v8f;

__device__ __forceinline__ int clamp_idx(int k, int lim) { return k < lim ? k : lim - 1; }

// ---- K1: zero output (used as agg accumulator), deg = 1.0 -----------------
__global__ __launch_bounds__(256) void tg_init_kernel(float* __restrict__ out, int out_n,
                                                      float* __restrict__ deg, int deg_n) {
    int i = blockIdx.x * blockDim.x + threadIdx.x;
    int stride = gridDim.x * blockDim.x;
    for (int j = i; j < out_n; j += stride) out[j] = 0.0f;
    for (int j = i; j < deg_n; j += stride) deg[j] = 1.0f;
}

// ---- K2: degree accumulation ----------------------------------------------
__global__ __launch_bounds__(256) void tg_degree_kernel(const int* __restrict__ col,
                                                        const float* __restrict__ ew,
                                                        float* __restrict__ deg, int E) {
    int e = blockIdx.x * blockDim.x + threadIdx.x;
    if (e < E) atomicAdd(&deg[col[e]], ew[e]);
}

// ---- K3: dinv = rsqrt(deg) in place ---------------------------------------
__global__ __launch_bounds__(256) void tg_rsqrt_kernel(float* __restrict__ deg, int n) {
    int i = blockIdx.x * blockDim.x + threadIdx.x;
    if (i < n) {
        float d = deg[i];
        deg[i] = (d > 0.0f) ? rsqrtf(d) : 0.0f;
    }
}

// ---- K4: edge scatter. Persistent wave32s, grid-stride over edges.
// Lane f handles features 2f,2f+1; its W columns are loop-invariant and
// hoisted into registers (one b64 load per k). Per-edge scalars load at
// uniform addresses (hardware broadcast) then readfirstlane-scalarized.
template <int IN_DIM>
__global__ __launch_bounds__(256) void tg_scatter_kernel(const int* __restrict__ row_idx,
                                                         const int* __restrict__ col_idx,
                                                         const float* __restrict__ ew,
                                                         const float* __restrict__ x,
                                                         const float* __restrict__ W,  // [IN_DIM,64]
                                                         const float* __restrict__ dinv,
                                                         float* __restrict__ agg,      // [N,64]
                                                         int E) {
    const int lane = threadIdx.x & 31;
    const int waveId = blockIdx.x * (blockDim.x >> 5) + (threadIdx.x >> 5);
    const int nWaves = gridDim.x * (blockDim.x >> 5);

    // Hoist this lane's two W columns (8B-aligned since feature index is even).
    const v2f* W2 = (const v2f*)W;  // [IN_DIM][32] of float2
    v2f wv[IN_DIM];
#pragma unroll
    for (int k = 0; k < IN_DIM; ++k) wv[k] = W2[k * 32 + lane];

    const int f0 = lane * 2;

    for (int e = waveId; e < E; e += nWaves) {
        int r = __builtin_amdgcn_readfirstlane(row_idx[e]);
        int c = __builtin_amdgcn_readfirstlane(col_idx[e]);
        float w = ew[e];
        float norm = dinv[r] * w * dinv[c];

        float h0 = 0.0f, h1 = 0.0f;
#pragma unroll
        for (int k = 0; k < IN_DIM; ++k) {
            float xk = x[r * IN_DIM + k];  // uniform address -> broadcast
            h0 = fmaf(xk, wv[k].x, h0);
            h1 = fmaf(xk, wv[k].y, h1);
        }
        float* dst = agg + (size_t)c * 64 + f0;
        atomicAdd(dst,     norm * h0);
        atomicAdd(dst + 1, norm * h1);
    }
}

// ---- K5: finalize. One wave32 per 16-node tile. h computed with
// V_WMMA_F32_16X16X4_F32 (A:16x4 f32, B:4x16 f32, D:16x16 f32), K padded to 4.
// VGPR layouts per CDNA5 ISA 7.12.2:
//   A: lanes 0-15 hold {K=0,K=1} for M=lane; lanes 16-31 hold {K=2,K=3}.
//   B: lanes 0-15 hold {K=0,K=1} for N=lane; lanes 16-31 hold {K=2,K=3}.
//   D: VGPR r = row (lane<16 ? r : 8+r), col = lane&15.
// K-padding: loads use clamped (always in-bounds) indices; the *value* is
// zeroed with a select, so no exec-mask branches are generated.
template <int IN_DIM>
__global__ __launch_bounds__(256) void tg_finalize_kernel(const float* __restrict__ x,    // [N,IN_DIM]
                                                          const float* __restrict__ W,    // [IN_DIM,64]
                                                          const float* __restrict__ bias, // [64]
                                                          const float* __restrict__ dinv, // [N]
                                                          float* __restrict__ out,        // [N,64] = agg in
                                                          int N) {
    const int lane = threadIdx.x & 31;
    const int tile = blockIdx.x * (blockDim.x >> 5) + (threadIdx.x >> 5);
    const int base = tile * 16;
    if (base >= N) return;  // uniform within the wave

    const int mA = lane & 15;
    const int kBase = (lane < 16) ? 0 : 2;
    const int mOff = (lane < 16) ? 0 : 8;
    const int k0 = clamp_idx(kBase + 0, IN_DIM);
    const int k1 = clamp_idx(kBase + 1, IN_DIM);
    const bool v0 = (kBase + 0) < IN_DIM;
    const bool v1 = (kBase + 1) < IN_DIM;

    if (base + 16 <= N) {
        // ---------------- full-tile fast path: branch-free ----------------
        const int nodeA = base + mA;
        float ax0 = x[(size_t)nodeA * IN_DIM + k0];
        float ax1 = x[(size_t)nodeA * IN_DIM + k1];
        v2f a;
        a.x = v0 ? ax0 : 0.0f;
        a.y = v1 ? ax1 : 0.0f;

        // dinv^2 for this lane-half's 8 rows (broadcast loads, hoisted).
        float di2[8];
#pragma unroll
        for (int r = 0; r < 8; ++r) {
            float di = dinv[base + mOff + r];
            di2[r] = di * di;
        }

#pragma unroll
        for (int n = 0; n < 4; ++n) {
            int feat = n * 16 + mA;
            float bx = W[k0 * 64 + feat];
            float by = W[k1 * 64 + feat];
            v2f b;
            b.x = v0 ? bx : 0.0f;
            b.y = v1 ? by : 0.0f;
            v8f c = {};
            // 8 args: (neg_a, A, neg_b, B, c_mod, C, reuse_a, reuse_b)
            c = __builtin_amdgcn_wmma_f32_16x16x4_f32(false, a, false, b, (short)0, c, false, false);

            float bf = bias[feat];
#pragma unroll
            for (int r = 0; r < 8; ++r) {
                float* p = out + (size_t)(base + mOff + r) * 64 + feat;
                float v = *p + di2[r] * c[r] + bf;
                *p = fmaxf(v, 0.0f);
            }
        }
    } else {
        // ------------- tail tile: clamped loads, guarded RMW --------------
        const int nodeA = clamp_idx(base + mA, N);  // in-bounds load address
        const bool rowA = (base + mA) < N;
        float ax0 = x[(size_t)nodeA * IN_DIM + k0];
        float ax1 = x[(size_t)nodeA * IN_DIM + k1];
        v2f a;
        a.x = (rowA && v0) ? ax0 : 0.0f;
        a.y = (rowA && v1) ? ax1 : 0.0f;

#pragma unroll
        for (int n = 0; n < 4; ++n) {
            int feat = n * 16 + mA;
            float bx = W[k0 * 64 + feat];
            float by = W[k1 * 64 + feat];
            v2f b;
            b.x = v0 ? bx : 0.0f;
            b.y = v1 ? by : 0.0f;
            v8f c = {};
            c = __builtin_amdgcn_wmma_f32_16x16x4_f32(false, a, false, b, (short)0, c, false, false);

            float bf = bias[feat];
#pragma unroll
            for (int r = 0; r < 8; ++r) {
                int node = base + mOff + r;
                if (node < N) {
                    float di = dinv[node];
                    float* p = out + (size_t)node * 64 + feat;
                    float v = *p + di * di * c[r] + bf;
                    *p = fmaxf(v, 0.0f);
                }
            }
        }
    }
}

extern "C" void kernel_launch(void* const* d_in, const int* in_sizes, int n_in,
                              void* d_out, int out_size, void* d_ws, size_t ws_size,
                              hipStream_t stream) {
    const float* x_z  = (const float*)d_in[0];
    const float* x_o  = (const float*)d_in[1];
    const float* x_g  = (const float*)d_in[2];
    const int*   ei_z = (const int*)d_in[3];
    const int*   ei_o = (const int*)d_in[4];
    const int*   ei_g = (const int*)d_in[5];
    const float* W_z  = (const float*)d_in[6];
    const float* W_o  = (const float*)d_in[7];
    const float* W_g  = (const float*)d_in[8];
    const float* b_z  = (const float*)d_in[9];
    const float* b_o  = (const float*)d_in[10];
    const float* b_g  = (const float*)d_in[11];
    const float* ew_z = (const float*)d_in[12];
    const float* ew_o = (const float*)d_in[13];
    const float* ew_g = (const float*)d_in[14];

    const int Nz = in_sizes[0] / 2;
    const int No = in_sizes[1] / 3;
    const int Ng = in_sizes[2] / 1;
    const int Ez = in_sizes[3] / 2;
    const int Eo = in_sizes[4] / 2;
    const int Eg = in_sizes[5] / 2;

    float* deg   = (float*)d_ws;       // dinv computed in place
    float* deg_z = deg;
    float* deg_o = deg + Nz;
    float* deg_g = deg + Nz + No;
    const int degN = Nz + No + Ng;

    float* out_z = (float*)d_out;
    float* out_o = out_z + (size_t)Nz * 64;
    float* out_g = out_o + (size_t)No * 64;

    // K1: zero agg (d_out) + deg = 1
    tg_init_kernel<<<2048, 256, 0, stream>>>((float*)d_out, out_size, deg, degN);

    // K2: degree accumulation (col = second row of edge_index)
    tg_degree_kernel<<<(Ez + 255) / 256, 256, 0, stream>>>(ei_z + Ez, ew_z, deg_z, Ez);
    tg_degree_kernel<<<(Eo + 255) / 256, 256, 0, stream>>>(ei_o + Eo, ew_o, deg_o, Eo);
    tg_degree_kernel<<<(Eg + 255) / 256, 256, 0, stream>>>(ei_g + Eg, ew_g, deg_g, Eg);

    // K3: dinv = rsqrt(deg)
    tg_rsqrt_kernel<<<(degN + 255) / 256, 256, 0, stream>>>(deg, degN);

    // K4: edge scatter. Persistent waves: cap the grid so each wave iterates
    // several edges and amortizes the hoisted W register loads.
    auto scatter_blocks = [](int E) {
        int b = (E + 7) / 8;          // 8 waves per 256-thread block
        return b < 32768 ? b : 32768; // ~262k waves max, grid-stride beyond
    };
    tg_scatter_kernel<2><<<scatter_blocks(Ez), 256, 0, stream>>>(ei_z, ei_z + Ez, ew_z, x_z, W_z, deg_z, out_z, Ez);
    tg_scatter_kernel<3><<<scatter_blocks(Eo), 256, 0, stream>>>(ei_o, ei_o + Eo, ew_o, x_o, W_o, deg_o, out_o, Eo);
    tg_scatter_kernel<1><<<scatter_blocks(Eg), 256, 0, stream>>>(ei_g, ei_g + Eg, ew_g, x_g, W_g, deg_g, out_g, Eg);

    // K5: WMMA self-loop + bias + relu (8 tiles/block = 128 nodes/block)
    tg_finalize_kernel<2><<<(Nz + 127) / 128, 256, 0, stream>>>(x_z, W_z, b_z, deg_z, out_z, Nz);
    tg_finalize_kernel<3><<<(No + 127) / 128, 256, 0, stream>>>(x_o, W_o, b_o, deg_o, out_o, No);
    tg_finalize_kernel<1><<<(Ng + 127) / 128, 256, 0, stream>>>(x_g, W_g, b_g, deg_g, out_g, Ng);
}